// E3nnSimpleEdgeBlock_11587821764686
// MI455X (gfx1250) — compile-verified
//
#include <hip/hip_runtime.h>

#define E_TOT 50000

typedef float v2f __attribute__((ext_vector_type(2)));
typedef float v4f __attribute__((ext_vector_type(4)));
typedef float v8f __attribute__((ext_vector_type(8)));

__device__ __forceinline__ v8f wmma4(v2f a, v2f b, v8f c) {
  // D = A(16x4 f32) * B(4x16 f32) + C(16x16 f32)
  return __builtin_amdgcn_wmma_f32_16x16x4_f32(false, a, false, b, (short)0, c,
                                               false, false);
}

__device__ __forceinline__ void async_b128(unsigned lds_byte_off, const void* g) {
  asm volatile("global_load_async_to_lds_b128 %0, %1, off"
               :: "v"(lds_byte_off),
                  "v"((unsigned long long)(size_t)g)
               : "memory");
}
__device__ __forceinline__ void wait_async0() {
  asm volatile("s_wait_asynccnt 0x0" ::: "memory");
}

// ---- W staging: global (row-major [uv][w]) -> LDS K-pair-interleaved
// LDS chunk layout: [pl (128 pairs)][col (32)][2]  (= 8192 floats / chunk of 8 u)
__device__ __forceinline__ void stage_load(const float* __restrict__ Wp, int u0,
                                           int tid, v4f G0[4], v4f G1[4]) {
#pragma unroll
  for (int i = 0; i < 4; ++i) {
    int it = tid + i * 256;
    int pl = it >> 3, cg = it & 7;
    const float* p = Wp + (size_t)(u0 * 16 + pl) * 64 + cg * 4;
    G0[i] = *(const v4f*)p;          // row 2*pg, cols cg*4..+3
    G1[i] = *(const v4f*)(p + 32);   // row 2*pg+1
  }
}
__device__ __forceinline__ void stage_store(float* __restrict__ Wb, int tid,
                                            const v4f G0[4], const v4f G1[4]) {
#pragma unroll
  for (int i = 0; i < 4; ++i) {
    int it = tid + i * 256;
    int pl = it >> 3, cg = it & 7;
    float* d = Wb + pl * 64 + cg * 8;
#pragma unroll
    for (int t = 0; t < 4; ++t) {
      v2f w;
      w.x = G0[i][t];
      w.y = G1[i][t];
      *(v2f*)(d + t * 2) = w;        // contiguous 32B -> mergeable ds stores
    }
  }
}

__device__ __forceinline__ void compute_chunk(
    int phase, int u0, const float* __restrict__ Wb,
    const float* __restrict__ rec, int h, int n,
    v8f (&acc0)[2], v8f (&acc1)[3][2]) {
  const float I3 = 0.57735026918962576451f;  // 1/sqrt(3)
  const float I6 = 0.40824829046386301637f;  // 1/sqrt(6)

  switch (phase) {
    case 0:  // W000: Z = a0[u]*b0[v]
      for (int uu = 0; uu < 8; ++uu) {
        float a = rec[u0 + uu];
#pragma unroll
        for (int q = 0; q < 8; ++q) {
          int v0 = q * 4 + h * 2;
          int pl = uu * 16 + q * 2 + h;
          v2f A;
          A.x = a * rec[32 + v0];
          A.y = a * rec[32 + v0 + 1];
#pragma unroll
          for (int nt = 0; nt < 2; ++nt) {
            v2f B = *(const v2f*)&Wb[pl * 64 + (nt * 16 + n) * 2];
            acc0[nt] = wmma4(A, B, acc0[nt]);
          }
        }
      }
      break;
    case 1:  // W110: Z = dot3(I3*a1[u], b1[v])  (scale folded into a1)
      for (int uu = 0; uu < 8; ++uu) {
        int u = u0 + uu;
        float ax = I3 * rec[64 + 3 * u], ay = I3 * rec[64 + 3 * u + 1],
              az = I3 * rec[64 + 3 * u + 2];
#pragma unroll
        for (int q = 0; q < 8; ++q) {
          int v0 = q * 4 + h * 2;
          int pl = uu * 16 + q * 2 + h;
          v2f A;
#pragma unroll
          for (int j = 0; j < 2; ++j) {
            const float* b = rec + 160 + 3 * (v0 + j);
            A[j] = ax * b[0] + ay * b[1] + az * b[2];
          }
#pragma unroll
          for (int nt = 0; nt < 2; ++nt) {
            v2f B = *(const v2f*)&Wb[pl * 64 + (nt * 16 + n) * 2];
            acc0[nt] = wmma4(A, B, acc0[nt]);
          }
        }
      }
      break;
    case 2:  // W011: Zk = (I3*a0[u])*b1[v,k]
      for (int uu = 0; uu < 8; ++uu) {
        float a = I3 * rec[u0 + uu];
#pragma unroll
        for (int q = 0; q < 8; ++q) {
          int v0 = q * 4 + h * 2;
          int pl = uu * 16 + q * 2 + h;
          v2f Ak[3];
#pragma unroll
          for (int j = 0; j < 2; ++j) {
            const float* b = rec + 160 + 3 * (v0 + j);
            Ak[0][j] = a * b[0];
            Ak[1][j] = a * b[1];
            Ak[2][j] = a * b[2];
          }
#pragma unroll
          for (int nt = 0; nt < 2; ++nt) {
            v2f B = *(const v2f*)&Wb[pl * 64 + (nt * 16 + n) * 2];
#pragma unroll
            for (int k = 0; k < 3; ++k)
              acc1[k][nt] = wmma4(Ak[k], B, acc1[k][nt]);
          }
        }
      }
      break;
    case 3:  // W101: Zk = (I3*a1[u,k])*b0[v]
      for (int uu = 0; uu < 8; ++uu) {
        int u = u0 + uu;
        float ax = I3 * rec[64 + 3 * u], ay = I3 * rec[64 + 3 * u + 1],
              az = I3 * rec[64 + 3 * u + 2];
#pragma unroll
        for (int q = 0; q < 8; ++q) {
          int v0 = q * 4 + h * 2;
          int pl = uu * 16 + q * 2 + h;
          v2f Ak[3];
#pragma unroll
          for (int j = 0; j < 2; ++j) {
            float b = rec[32 + v0 + j];
            Ak[0][j] = ax * b;
            Ak[1][j] = ay * b;
            Ak[2][j] = az * b;
          }
#pragma unroll
          for (int nt = 0; nt < 2; ++nt) {
            v2f B = *(const v2f*)&Wb[pl * 64 + (nt * 16 + n) * 2];
#pragma unroll
            for (int k = 0; k < 3; ++k)
              acc1[k][nt] = wmma4(Ak[k], B, acc1[k][nt]);
          }
        }
      }
      break;
    default:  // W111: Zk = cross(I6*a1[u], b1[v])_k  (scale folded into a1)
      for (int uu = 0; uu < 8; ++uu) {
        int u = u0 + uu;
        float ax = I6 * rec[64 + 3 * u], ay = I6 * rec[64 + 3 * u + 1],
              az = I6 * rec[64 + 3 * u + 2];
#pragma unroll
        for (int q = 0; q < 8; ++q) {
          int v0 = q * 4 + h * 2;
          int pl = uu * 16 + q * 2 + h;
          v2f Ak[3];
#pragma unroll
          for (int j = 0; j < 2; ++j) {
            const float* b = rec + 160 + 3 * (v0 + j);
            Ak[0][j] = ay * b[2] - az * b[1];
            Ak[1][j] = az * b[0] - ax * b[2];
            Ak[2][j] = ax * b[1] - ay * b[0];
          }
#pragma unroll
          for (int nt = 0; nt < 2; ++nt) {
            v2f B = *(const v2f*)&Wb[pl * 64 + (nt * 16 + n) * 2];
#pragma unroll
            for (int k = 0; k < 3; ++k)
              acc1[k][nt] = wmma4(Ak[k], B, acc1[k][nt]);
          }
        }
      }
      break;
  }
}

__global__ __launch_bounds__(256) void e3nn_tp_kernel(
    const float* __restrict__ x1, const float* __restrict__ x2,
    const float* __restrict__ W000, const float* __restrict__ W110,
    const float* __restrict__ W011, const float* __restrict__ W101,
    const float* __restrict__ W111, float* __restrict__ out) {
  extern __shared__ float smem[];
  // smem[0..8192)      : W chunk buffer 0 (32 KB)
  // smem[8192..16384)  : W chunk buffer 1 (32 KB)
  // smem[16384..49152) : x tile, 128 edges * 256 floats (128 KB)
  float* Xl = smem + 16384;
  const unsigned X_LDS_B = 16384u * 4u;

  const int tid  = threadIdx.x;
  const int lane = tid & 31;
  const int wv   = tid >> 5;   // wave id 0..7, one 16-edge tile each
  const int h    = lane >> 4;  // half-wave: A/B K-group, C/D row group
  const int n    = lane & 15;  // N column

  const float C0 = 0.02209708691207961102f;  // 1/sqrt(2048)
  const float C1 = 0.03125f;                 // 1/sqrt(1024)

  // ---- stage x1/x2 for 128 edges into LDS (async), record layout per edge:
  // [0:32) a0 | [32:64) b0 | [64:160) a1 (u-major xyz) | [160:256) b1
  for (int m = tid; m < 8192; m += 256) {
    int s = m >> 12;              // 0: x1, 1: x2
    int r = m & 4095;
    int e = r >> 5;               // local edge 0..127
    int q = r & 31;               // 16-byte chunk in 128-float row
    long long eg = (long long)blockIdx.x * 128 + e;
    if (eg < E_TOT) {
      const float* src = (s ? x2 : x1) + eg * 128 + q * 4;
      int f0 = q * 4;
      int ro = (f0 < 32) ? (f0 + s * 32) : ((s ? 160 : 64) + (f0 - 32));
      async_b128(X_LDS_B + (unsigned)(e * 256 + ro) * 4u, src);
    }
  }

  const int eBase = blockIdx.x * 128 + wv * 16;
  const bool active = eBase < E_TOT;  // E % 16 == 0 -> tiles all-or-nothing
  const float* rec = Xl + (wv * 16 + (lane & 15)) * 256;

  v8f acc0[2] = {};
  v8f acc1[3][2] = {};

  v4f G0[4], G1[4];

  // prime the pipeline: chunk 0 -> buffer 0
  stage_load(W000, 0, tid, G0, G1);
  stage_store(smem, tid, G0, G1);
  wait_async0();
  __syncthreads();

  // 20 chunks: phase = c/4 (W tensor), u0 = (c%4)*8
  for (int c = 0; c < 20; ++c) {
    if (c < 19) {
      int cn = c + 1;
      int ph = cn >> 2;
      const float* Wp = (ph == 0) ? W000
                       : (ph == 1) ? W110
                       : (ph == 2) ? W011
                       : (ph == 3) ? W101 : W111;
      stage_load(Wp, (cn & 3) * 8, tid, G0, G1);
    }
    if (active)
      compute_chunk(c >> 2, (c & 3) * 8, smem + (c & 1) * 8192, rec, h, n,
                    acc0, acc1);
    if (c < 19) stage_store(smem + ((c + 1) & 1) * 8192, tid, G0, G1);
    __syncthreads();
  }

  // ---- epilogue: scale + store
  if (active) {
#pragma unroll
    for (int r = 0; r < 8; ++r) {
      float* orow = out + (size_t)(eBase + r + 8 * h) * 128;
#pragma unroll
      for (int nt = 0; nt < 2; ++nt) {
        int ch = nt * 16 + n;
        orow[ch] = C0 * acc0[nt][r];
#pragma unroll
        for (int k = 0; k < 3; ++k)
          orow[32 + ch * 3 + k] = C1 * acc1[k][nt][r];
      }
    }
  }
}

extern "C" void kernel_launch(void* const* d_in, const int* in_sizes, int n_in,
                              void* d_out, int out_size, void* d_ws, size_t ws_size,
                              hipStream_t stream) {
  const float* x1   = (const float*)d_in[0];
  const float* x2   = (const float*)d_in[1];
  const float* W000 = (const float*)d_in[2];
  const float* W110 = (const float*)d_in[3];
  const float* W011 = (const float*)d_in[4];
  const float* W101 = (const float*)d_in[5];
  const float* W111 = (const float*)d_in[6];
  float* out = (float*)d_out;

  dim3 grid((E_TOT + 127) / 128);   // 128 edges per workgroup (8 waves x 16)
  dim3 block(256);
  size_t shmem = (16384 + 128 * 256) * sizeof(float);  // 192 KB (<= 320 KB/WGP)
  hipLaunchKernelGGL(e3nn_tp_kernel, grid, block, shmem, stream,
                     x1, x2, W000, W110, W011, W101, W111, out);
}